// Voxelizer_6708738916535
// MI455X (gfx1250) — compile-verified
//
#include <hip/hip_runtime.h>
#include <hip/hip_bf16.h>

// Voxelizer as CP-decomposition GEMMs on CDNA5 WMMA (wave32).
//
// out[b,ch,x,y,z] = sum_a [ch_a==ch] * Wx[a,x] * Wy[a,y] * Wz[a,z]
// where W*[a,p] = exp(-(0.5*RES^2/r^2) * (scaled_* - (cell+0.5))^2) if the
// cell is within CUBES(=5) of the atom's base voxel, else 0.
//
// Per (b,ch,y) slice: out[x,z] = A(32x64) * B(64x32) with
// V_WMMA_F32_16X16X4_F32 (2x2 tiles of 16x16, 16 K-chunks of 4 atoms).
// A folds the channel mask and Wy; B = Wz so the GEMM N dimension is z,
// the innermost (contiguous) output axis -> coalesced global stores
// (output write traffic is the roofline floor: 5.24 MB @ 23.3 TB/s).
//
// The two reference "dumb" pin atoms land >5 cells outside the 32^3 crop
// (base cells 6 and 166 vs crop 70..101) so they are provably zero here.

typedef __attribute__((ext_vector_type(2))) float v2f;
typedef __attribute__((ext_vector_type(8))) float v8f;

#define NB    8
#define NA    64          // real atoms (dummies contribute 0 to the crop)
#define NCH   5
#define CROP  32
#define LO    70          // D=173, center 86, crop 70..101
#define KCH   (NA / 4)    // 16 K-chunks of 4 for wmma 16x16x4

__global__ __launch_bounds__(256) void voxelizer_wmma_kernel(
    const float* __restrict__ coords,    // [NB, NA, 3]
    const int*   __restrict__ chan,      // [NB, NA]
    const float* __restrict__ radius,    // [NB, NA]
    float*       __restrict__ out)       // [NB, NCH, 32, 32, 32]
{
    // LDS: separable axis profiles, [axis][atom][crop-pos]
    __shared__ float W[3 * NA * CROP];   // 24 KB
    __shared__ int   chs[NA];

    const int b   = blockIdx.x;
    const int tid = threadIdx.x;

    if (tid < NA) chs[tid] = chan[b * NA + tid];

    // ---- Phase 1: build Wx/Wy/Wz profiles cooperatively (6144 entries) ----
    for (int idx = tid; idx < 3 * NA * CROP; idx += 256) {
        const int axis = idx / (NA * CROP);
        const int rem  = idx - axis * (NA * CROP);
        const int a    = rem >> 5;        // atom
        const int x    = rem & 31;        // crop-local position
        const float c  = coords[(b * NA + a) * 3 + axis];
        const float r  = radius[b * NA + a];
        // scaled = (c + BOX)/RES + (CUBES+1); /0.25 == *4.0 exactly
        const float scaled = (c + 20.0f) * 4.0f + 6.0f;
        const int   base   = (int)floorf(scaled);
        const int   g      = LO + x;      // global cell index
        const int   dcell  = g - base;
        float w = 0.0f;
        if (dcell >= -5 && dcell <= 5) {
            const float d    = scaled - ((float)g + 0.5f);
            const float coef = 0.03125f / (r * r);   // 0.5 * RES^2 / r^2
            w = expf(-coef * d * d);
        }
        W[idx] = w;
    }
    __syncthreads();

    // ---- Phase 2: WMMA GEMMs, one (ch,y) slice at a time per wave ----
    // readfirstlane makes the wave id an SGPR value: the slice loop becomes
    // a uniform scalar loop (no exec-mask divergence around the WMMAs, which
    // require EXEC == all ones).
    const int wave = __builtin_amdgcn_readfirstlane(tid >> 5);
    const int lane = tid & 31;
    const int i    = lane & 15;   // M-index for A operand, N-index for B / C/D
    const int h    = lane >> 4;   // lane half: selects K pair / M+8 in C/D

    const float* Wx = &W[0];
    const float* Wy = &W[NA * CROP];
    const float* Wz = &W[2 * NA * CROP];

    for (int s = wave; s < NCH * CROP; s += 8) {
        const int ch = s >> 5;    // channel
        const int y  = s & 31;    // fixed y for this slice

        // acc[tx][tz]: tx tiles x (GEMM M), tz tiles z (GEMM N, contiguous)
        v8f acc00 = {}, acc01 = {}, acc10 = {}, acc11 = {};

        for (int kc = 0; kc < KCH; ++kc) {
            // k-slot -> atom mapping (identical for A and B operands):
            // k = 2*h + v  ->  a = 4*kc + k
            const int a0 = kc * 4 + 2 * h;
            const int a1 = a0 + 1;
            // fold channel mask and Wy into the A operand
            const float m0 = (chs[a0] == ch) ? Wy[a0 * 32 + y] : 0.0f;
            const float m1 = (chs[a1] == ch) ? Wy[a1 * 32 + y] : 0.0f;

            v2f A0, A1, B0, B1;
            A0[0] = m0 * Wx[a0 * 32 + i];        A0[1] = m1 * Wx[a1 * 32 + i];
            A1[0] = m0 * Wx[a0 * 32 + 16 + i];   A1[1] = m1 * Wx[a1 * 32 + 16 + i];
            B0[0] = Wz[a0 * 32 + i];             B0[1] = Wz[a1 * 32 + i];
            B1[0] = Wz[a0 * 32 + 16 + i];        B1[1] = Wz[a1 * 32 + 16 + i];

            // 8 args: (neg_a, A, neg_b, B, c_mod, C, reuse_a, reuse_b)
            acc00 = __builtin_amdgcn_wmma_f32_16x16x4_f32(false, A0, false, B0,
                                                          (short)0, acc00, false, false);
            acc01 = __builtin_amdgcn_wmma_f32_16x16x4_f32(false, A0, false, B1,
                                                          (short)0, acc01, false, false);
            acc10 = __builtin_amdgcn_wmma_f32_16x16x4_f32(false, A1, false, B0,
                                                          (short)0, acc10, false, false);
            acc11 = __builtin_amdgcn_wmma_f32_16x16x4_f32(false, A1, false, B1,
                                                          (short)0, acc11, false, false);
        }

        // C/D layout: VGPR r, lane-half h -> M(x) = r + 8h ; N(z) = i.
        // z is innermost in memory: each store writes two contiguous 64B runs.
        float* op = out + ((size_t)(b * NCH + ch) * (CROP * CROP * CROP)) + y * 32;
        #pragma unroll
        for (int r = 0; r < 8; ++r) {
            const int x = r + 8 * h;
            op[(size_t)(x)      * 1024 + (i)     ] = acc00[r];
            op[(size_t)(x)      * 1024 + (16 + i)] = acc01[r];
            op[(size_t)(16 + x) * 1024 + (i)     ] = acc10[r];
            op[(size_t)(16 + x) * 1024 + (16 + i)] = acc11[r];
        }
    }
}

extern "C" void kernel_launch(void* const* d_in, const int* in_sizes, int n_in,
                              void* d_out, int out_size, void* d_ws, size_t ws_size,
                              hipStream_t stream) {
    const float* coords = (const float*)d_in[0];   // [8,64,3] f32
    const int*   chan   = (const int*)  d_in[1];   // [8,64]   i32
    const float* radius = (const float*)d_in[2];   // [8,64]   f32
    float*       out    = (float*)d_out;           // [8,5,32,32,32] f32

    // One workgroup per batch; 8 wave32 waves each; every output voxel is
    // written exactly once (no pre-zero, no atomics, deterministic).
    voxelizer_wmma_kernel<<<NB, 256, 0, stream>>>(coords, chan, radius, out);
}